// GCNLayer_70858370449777
// MI455X (gfx1250) — compile-verified
//
#include <hip/hip_runtime.h>
#include <hip/hip_bf16.h>
#include <math.h>

#define N_NODES 50000
#define N_EDGES 800000
#define D 64
#define NEG_SLOPE 0.2f

typedef __attribute__((ext_vector_type(2))) float v2f;
typedef __attribute__((ext_vector_type(8))) float v8f;

// ------------------------------------------------------------------
// Kernel 1: h = feat @ W^T   via V_WMMA_F32_16X16X4_F32 (full f32).
// One wave32 per 16x16 output tile; K=64 consumed in 16 WMMA steps.
// A (16x4 f32):  lanes 0-15 -> M=lane, v0=K0 v1=K1 ; lanes 16-31 -> v0=K2 v1=K3
// B (4x16 f32):  B[k][n] = W[n][k]  -> same per-lane addressing pattern on W rows
// C/D (16x16 f32): VGPR r, lanes 0-15 -> M=r, N=lane ; lanes 16-31 -> M=r+8
// ------------------------------------------------------------------
__global__ __launch_bounds__(256) void k_gemm_wmma(const float* __restrict__ feat,
                                                   const float* __restrict__ W,
                                                   float* __restrict__ h) {
  const int wave = (blockIdx.x * blockDim.x + threadIdx.x) >> 5;
  const int lane = threadIdx.x & 31;
  const int tiles_n = D / 16;                    // 4
  const int tiles_m = N_NODES / 16;              // 3125 (exact)
  if (wave >= tiles_m * tiles_n) return;         // wave-uniform: EXEC stays all-1s
  const int tm = wave / tiles_n;
  const int tn = wave % tiles_n;
  const int hi  = lane >> 4;                     // 0 = low half-wave, 1 = high
  const int l16 = lane & 15;

  const float* arow = feat + (size_t)(tm * 16 + l16) * D;  // A row M=l16
  const float* brow = W    + (size_t)(tn * 16 + l16) * D;  // W row n -> B column n

  v8f c = {};
#pragma unroll
  for (int k0 = 0; k0 < D; k0 += 4) {
    v2f a, b;
    a.x = arow[k0 + 2 * hi + 0];
    a.y = arow[k0 + 2 * hi + 1];
    b.x = brow[k0 + 2 * hi + 0];
    b.y = brow[k0 + 2 * hi + 1];
    // 8 args: (neg_a, A, neg_b, B, c_mod, C, reuse_a, reuse_b)
    c = __builtin_amdgcn_wmma_f32_16x16x4_f32(false, a, false, b, (short)0, c,
                                              false, false);
  }
#pragma unroll
  for (int r = 0; r < 8; ++r) {
    h[(size_t)(tm * 16 + r + hi * 8) * D + tn * 16 + l16] = c[r];
  }
}

// ------------------------------------------------------------------
// Kernel 2: per-call init (harness poisons buffers; we must zero ours).
// ------------------------------------------------------------------
__global__ void k_init(float* __restrict__ hagg, float* __restrict__ emax,
                       float* __restrict__ denom) {
  int i = blockIdx.x * blockDim.x + threadIdx.x;
  if (i < N_NODES * D) hagg[i] = 0.0f;
  if (i < N_NODES) {
    emax[i]  = -INFINITY;
    denom[i] = 0.0f;
  }
}

// ------------------------------------------------------------------
// Kernel 3: h_agg[dst] += h[src]  (segment_sum over edges).
// One thread per (edge, 4-float quad): float4 gather + 4 f32 atomics.
// ------------------------------------------------------------------
__global__ void k_scatter(const float* __restrict__ h,
                          const int* __restrict__ src,
                          const int* __restrict__ dst,
                          float* __restrict__ hagg) {
  int t = blockIdx.x * blockDim.x + threadIdx.x;
  if (t >= N_EDGES * (D / 4)) return;
  int e = t >> 4;              // D/4 == 16 quads per edge
  int d = (t & 15) * 4;
  int s  = src[e];
  int dd = dst[e];
  const float4 v = *(const float4*)(h + (size_t)s * D + d);
  float* out = hagg + (size_t)dd * D + d;
  atomicAdd(out + 0, v.x);
  atomicAdd(out + 1, v.y);
  atomicAdd(out + 2, v.z);
  atomicAdd(out + 3, v.w);
}

// ------------------------------------------------------------------
// Kernel 4: e = leaky_relu( sum_d h_agg[src,d] * tanh(h_agg[dst,d]) )
// One wave32 per edge; 2 features/lane; butterfly shfl_xor reduction.
// Also running segment_max via float atomicMax (global_atomic_max_num_f32).
// ------------------------------------------------------------------
__global__ __launch_bounds__(256) void k_edge(const float* __restrict__ hagg,
                                              const int* __restrict__ src,
                                              const int* __restrict__ dst,
                                              float* __restrict__ e_out,
                                              float* __restrict__ emax) {
  int wave = (blockIdx.x * blockDim.x + threadIdx.x) >> 5;
  int lane = threadIdx.x & 31;
  if (wave >= N_EDGES) return;
  int s = src[wave];
  int d = dst[wave];
  const float* hs = hagg + (size_t)s * D;
  const float* hd = hagg + (size_t)d * D;
  float p = hs[lane] * tanhf(hd[lane]) + hs[lane + 32] * tanhf(hd[lane + 32]);
#pragma unroll
  for (int m = 16; m >= 1; m >>= 1) p += __shfl_xor(p, m, 32);
  if (lane == 0) {
    float ev = (p >= 0.0f) ? p : NEG_SLOPE * p;
    e_out[wave] = ev;
    atomicMax(emax + d, ev);
  }
}

// ------------------------------------------------------------------
// Kernel 5: e_exp = exp(e - emax[dst]); denom[dst] += e_exp
// ------------------------------------------------------------------
__global__ void k_exp(const float* __restrict__ e, const int* __restrict__ dst,
                      const float* __restrict__ emax, float* __restrict__ eexp,
                      float* __restrict__ denom) {
  int i = blockIdx.x * blockDim.x + threadIdx.x;
  if (i >= N_EDGES) return;
  int d = dst[i];
  float v = expf(e[i] - emax[d]);
  eexp[i] = v;
  atomicAdd(denom + d, v);
}

// ------------------------------------------------------------------
// Kernel 6: e_soft = e_exp / denom[dst]
// ------------------------------------------------------------------
__global__ void k_norm(const float* __restrict__ eexp, const int* __restrict__ dst,
                       const float* __restrict__ denom, float* __restrict__ esoft) {
  int i = blockIdx.x * blockDim.x + threadIdx.x;
  if (i >= N_EDGES) return;
  esoft[i] = eexp[i] / denom[dst[i]];
}

extern "C" void kernel_launch(void* const* d_in, const int* in_sizes, int n_in,
                              void* d_out, int out_size, void* d_ws, size_t ws_size,
                              hipStream_t stream) {
  const float* feat = (const float*)d_in[0];
  const float* W    = (const float*)d_in[1];
  const int*   src  = (const int*)d_in[2];
  const int*   dst  = (const int*)d_in[3];

  float* out   = (float*)d_out;
  float* hagg  = out;                               // [N*D] output 0
  float* esoft = out + (size_t)N_NODES * D;         // [E]   output 1

  float* ws    = (float*)d_ws;
  float* h     = ws;                                // [N*D]
  float* e     = h + (size_t)N_NODES * D;           // [E]
  float* eexp  = e + N_EDGES;                       // [E]
  float* emax  = eexp + N_EDGES;                    // [N]
  float* denom = emax + N_NODES;                    // [N]

  const int waves = (N_NODES / 16) * (D / 16);      // 12500 WMMA tiles
  k_gemm_wmma<<<(waves * 32 + 255) / 256, 256, 0, stream>>>(feat, W, h);
  k_init<<<(N_NODES * D + 255) / 256, 256, 0, stream>>>(hagg, emax, denom);
  k_scatter<<<(N_EDGES * (D / 4) + 255) / 256, 256, 0, stream>>>(h, src, dst, hagg);
  k_edge<<<(N_EDGES * 32 + 255) / 256, 256, 0, stream>>>(hagg, src, dst, e, emax);
  k_exp<<<(N_EDGES + 255) / 256, 256, 0, stream>>>(e, dst, emax, eexp, denom);
  k_norm<<<(N_EDGES + 255) / 256, 256, 0, stream>>>(eexp, dst, denom, esoft);
}